// GPT2AttentionWithCache_74706661147056
// MI455X (gfx1250) — compile-verified
//
#include <hip/hip_runtime.h>
#include <math.h>

typedef __attribute__((ext_vector_type(2))) float v2f;
typedef __attribute__((ext_vector_type(4))) float v4f;
typedef __attribute__((ext_vector_type(8))) float v8f;

#define B_      16
#define H_      2048
#define NH_     16
#define HD_     128
#define S_      4096
#define QKV_    (3 * H_)
#define FF_     (4 * H_)
#define KSPLIT  16
#define NCHUNK  16
#define SCHUNK  (S_ / NCHUNK)   // 256
#define PSTRIDE (HD_ + 4)       // 4-float header (m, l) + 128 acc

__device__ __forceinline__ float waveReduceSum(float v) {
#pragma unroll
  for (int m = 16; m > 0; m >>= 1) v += __shfl_xor(v, m, 32);
  return v;
}
__device__ __forceinline__ float waveReduceMax(float v) {
#pragma unroll
  for (int m = 16; m > 0; m >>= 1) v = fmaxf(v, __shfl_xor(v, m, 32));
  return v;
}

// ---------------- LayerNorm: one block per batch row -----------------------
__global__ __launch_bounds__(256)
void ln_kernel(const float* __restrict__ x, const float* __restrict__ g,
               const float* __restrict__ bb, float* __restrict__ out) {
  const int b = blockIdx.x;
  const float* row = x + (size_t)b * H_;
  float s = 0.f, ss = 0.f;
  for (int i = threadIdx.x; i < H_; i += 256) {
    float v = row[i]; s += v; ss += v * v;
  }
  s = waveReduceSum(s); ss = waveReduceSum(ss);
  __shared__ float rs[8], rss[8], stats[2];
  const int wave = threadIdx.x >> 5, lane = threadIdx.x & 31;
  if (lane == 0) { rs[wave] = s; rss[wave] = ss; }
  __syncthreads();
  if (threadIdx.x == 0) {
    float S = 0.f, SS = 0.f;
#pragma unroll
    for (int w = 0; w < 8; ++w) { S += rs[w]; SS += rss[w]; }
    const float mu = S * (1.0f / H_);
    const float var = SS * (1.0f / H_) - mu * mu;
    stats[0] = mu; stats[1] = rsqrtf(var + 1e-5f);
  }
  __syncthreads();
  const float mu = stats[0], rstd = stats[1];
  for (int i = threadIdx.x; i < H_; i += 256)
    out[(size_t)b * H_ + i] = (row[i] - mu) * rstd * g[i] + bb[i];
}

// ------- GEMM: C[16xN] = A[16xK] @ W[KxN], split-K partials via WMMA -------
// One wave computes one 16x16 tile via V_WMMA_F32_16X16X4_F32.
// A chunk staged in LDS; W streamed non-temporally (read-once).
__global__ __launch_bounds__(256)
void gemm16_partial(const float* __restrict__ A, const float* __restrict__ W,
                    float* __restrict__ Cpart, int K, int N) {
  const int kLen = K / KSPLIT;              // 128 or 512
  const int k0 = blockIdx.y * kLen;
  __shared__ float sA[16 * 512];            // 32 KB max
  for (int idx = threadIdx.x; idx < 16 * kLen; idx += 256) {
    const int r = idx / kLen, j = idx - r * kLen;
    sA[r * kLen + j] = A[(size_t)r * K + k0 + j];
  }
  __syncthreads();
  const int wave = threadIdx.x >> 5;
  const int lane = threadIdx.x & 31;
  const int n0 = (blockIdx.x * 8 + wave) * 16;
  const int rowA = lane & 15;               // M row held by this lane
  const int kOff = (lane >> 4) << 1;        // lanes 16-31 hold K+2,K+3
  const float* __restrict__ sArow = sA + rowA * kLen + kOff;
  const float* __restrict__ Wp = W + (size_t)(k0 + kOff) * N + n0 + (lane & 15);
  v8f acc = {};
#pragma unroll 4
  for (int kk = 0; kk < kLen; kk += 4) {
    v2f a;
    a.x = sArow[kk];
    a.y = sArow[kk + 1];
    v2f bm;
    bm.x = __builtin_nontemporal_load(Wp + (size_t)kk * N);
    bm.y = __builtin_nontemporal_load(Wp + (size_t)(kk + 1) * N);
    acc = __builtin_amdgcn_wmma_f32_16x16x4_f32(false, a, false, bm,
                                                (short)0, acc, false, false);
  }
  // C/D layout: vgpr j, lanes 0-15 -> M=j ; lanes 16-31 -> M=8+j ; N = lane%16
  const int rbase = (lane >> 4) << 3;
  float* out = Cpart + (size_t)blockIdx.y * 16 * N + n0 + (lane & 15);
#pragma unroll
  for (int j = 0; j < 8; ++j)
    out[(size_t)(rbase + j) * N] = acc[j];
}

// epi: 0=none, 1=+resid, 2=gelu_new
__global__ __launch_bounds__(256)
void gemm16_finish(const float* __restrict__ Cpart,
                   const float* __restrict__ bias,
                   const float* __restrict__ resid,
                   float* __restrict__ out, int N, int epi) {
  const int idx = blockIdx.x * 256 + threadIdx.x;
  if (idx >= 16 * N) return;
  float s = 0.f;
#pragma unroll 4
  for (int c = 0; c < KSPLIT; ++c) s += Cpart[(size_t)c * 16 * N + idx];
  s += bias[idx % N];
  if (epi == 1) {
    s += resid[idx];
  } else if (epi == 2) {
    const float x = s;
    const float inner = 0.7978845608028654f * (x + 0.044715f * x * x * x);
    s = 0.5f * x * (1.0f + tanhf(inner));
  }
  out[idx] = s;
}

// ---------------- Flash-decode attention: partial per S-chunk --------------
__global__ __launch_bounds__(256)
void attn_partial(const float* __restrict__ qkv,
                  const float* __restrict__ Kc,
                  const float* __restrict__ Vc,
                  float* __restrict__ part) {
  const int bh = blockIdx.x;               // b*16 + h
  const int b = bh >> 4, h = bh & 15;
  const int c = blockIdx.y;                // S chunk
  const float scale = 0.08838834764831845f;  // 1/sqrt(128)
  __shared__ float sQ[HD_];
  __shared__ float sS[SCHUNK];
  __shared__ float red[8];
  __shared__ float sML[2];
  __shared__ float sAcc[HD_];
  const float* q = qkv + (size_t)b * QKV_ + h * HD_;
  if (threadIdx.x < HD_) sQ[threadIdx.x] = q[threadIdx.x];
  __syncthreads();
  const int wave = threadIdx.x >> 5, lane = threadIdx.x & 31;
  const v4f qv = ((const v4f*)sQ)[lane];
  const float* Kbase = Kc + ((size_t)bh * S_ + (size_t)c * SCHUNK) * HD_;
  // each wave: 32 rows; one 512B row per iteration, fully coalesced, NT
  for (int r = wave * 32; r < wave * 32 + 32; ++r) {
    const v4f kv =
        __builtin_nontemporal_load(((const v4f*)(Kbase + (size_t)r * HD_)) + lane);
    float d = kv.x * qv.x + kv.y * qv.y + kv.z * qv.z + kv.w * qv.w;
    d = waveReduceSum(d);
    if (lane == 0) sS[r] = d * scale;
  }
  __syncthreads();
  const float sc = sS[threadIdx.x];
  float m = waveReduceMax(sc);
  if (lane == 0) red[wave] = m;
  __syncthreads();
  if (threadIdx.x == 0) {
    float M = red[0];
#pragma unroll
    for (int w = 1; w < 8; ++w) M = fmaxf(M, red[w]);
    sML[0] = M;
  }
  __syncthreads();
  const float M = sML[0];
  const float p = __expf(sc - M);
  const float l = waveReduceSum(p);
  if (lane == 0) red[wave] = l;
  sS[threadIdx.x] = p;                     // own slot only
  __syncthreads();
  if (threadIdx.x == 0) {
    float L = 0.f;
#pragma unroll
    for (int w = 0; w < 8; ++w) L += red[w];
    sML[1] = L;
  }
  // weighted V: threads split 2x128 rows x 128 dims, coalesced NT streams
  const int d = threadIdx.x & 127;
  const int r0 = (threadIdx.x >> 7) * 128;
  const float* Vbase = Vc + ((size_t)bh * S_ + (size_t)c * SCHUNK + r0) * HD_;
  float acc = 0.f;
#pragma unroll 4
  for (int r = 0; r < 128; ++r)
    acc += sS[r0 + r] * __builtin_nontemporal_load(Vbase + (size_t)r * HD_ + d);
  if (threadIdx.x >= 128) sAcc[d] = acc;
  __syncthreads();
  float* pp = part + ((size_t)bh * NCHUNK + c) * PSTRIDE;
  if (threadIdx.x < 128) pp[4 + d] = acc + sAcc[d];
  if (threadIdx.x == 0) { pp[0] = sML[0]; pp[1] = sML[1]; }
}

// Combine chunk partials + the freshly-computed token's k/v (softmax over S+1)
__global__ __launch_bounds__(128)
void attn_combine(const float* __restrict__ part,
                  const float* __restrict__ qkv,
                  float* __restrict__ attn_out) {
  const int bh = blockIdx.x;
  const int b = bh >> 4, h = bh & 15;
  const int d = threadIdx.x;               // 0..127
  const float scale = 0.08838834764831845f;
  const float* q  = qkv + (size_t)b * QKV_ + h * HD_;
  const float* kn = qkv + (size_t)b * QKV_ + H_ + h * HD_;
  const float* vn = qkv + (size_t)b * QKV_ + 2 * H_ + h * HD_;
  __shared__ float red[4];
  __shared__ float sS;
  float t = q[d] * kn[d];
  t = waveReduceSum(t);
  const int wave = d >> 5, lane = d & 31;
  if (lane == 0) red[wave] = t;
  __syncthreads();
  if (d == 0) sS = (red[0] + red[1] + red[2] + red[3]) * scale;
  __syncthreads();
  const float sNew = sS;
  const float* pp = part + (size_t)bh * NCHUNK * PSTRIDE;
  float m = sNew;
#pragma unroll
  for (int c = 0; c < NCHUNK; ++c) m = fmaxf(m, pp[c * PSTRIDE]);
  const float wNew = __expf(sNew - m);
  float L = wNew;
  float acc = wNew * vn[d];
#pragma unroll
  for (int c = 0; c < NCHUNK; ++c) {
    const float w = __expf(pp[c * PSTRIDE] - m);
    L += pp[c * PSTRIDE + 1] * w;
    acc += pp[c * PSTRIDE + 4 + d] * w;
  }
  attn_out[(size_t)b * H_ + h * HD_ + d] = acc / L;
}

extern "C" void kernel_launch(void* const* d_in, const int* in_sizes, int n_in,
                              void* d_out, int out_size, void* d_ws, size_t ws_size,
                              hipStream_t stream) {
  (void)in_sizes; (void)n_in; (void)out_size; (void)ws_size;
  const float* hidden = (const float*)d_in[0];
  const float* Kc     = (const float*)d_in[1];
  const float* Vc     = (const float*)d_in[2];
  const float* ln1_g  = (const float*)d_in[3];
  const float* ln1_b  = (const float*)d_in[4];
  const float* W_qkv  = (const float*)d_in[5];
  const float* b_qkv  = (const float*)d_in[6];
  const float* W_proj = (const float*)d_in[7];
  const float* b_proj = (const float*)d_in[8];
  const float* ln2_g  = (const float*)d_in[9];
  const float* ln2_b  = (const float*)d_in[10];
  const float* W_fc   = (const float*)d_in[11];
  const float* b_fc   = (const float*)d_in[12];
  const float* W_out  = (const float*)d_in[13];
  const float* b_out  = (const float*)d_in[14];
  float* out = (float*)d_out;

  float* ws = (float*)d_ws;
  size_t o = 0;
  float* x_ln  = ws + o; o += (size_t)B_ * H_;           // 16x2048
  float* qkv   = ws + o; o += (size_t)B_ * QKV_;         // 16x6144
  float* h     = ws + o; o += (size_t)B_ * H_;
  float* y_ln  = ws + o; o += (size_t)B_ * H_;
  float* ff    = ws + o; o += (size_t)B_ * FF_;          // 16x8192
  float* attn  = ws + o; o += (size_t)B_ * H_;
  float* part  = ws + o; o += (size_t)B_ * NH_ * NCHUNK * PSTRIDE;
  float* cpart = ws + o;                                  // KSPLIT*16*8192 max

  const dim3 blk(256);
  // x = LN1(hidden)
  ln_kernel<<<B_, blk, 0, stream>>>(hidden, ln1_g, ln1_b, x_ln);
  // qkv = x @ W_qkv + b_qkv
  gemm16_partial<<<dim3(QKV_ / 128, KSPLIT), blk, 0, stream>>>(x_ln, W_qkv, cpart, H_, QKV_);
  gemm16_finish<<<(B_ * QKV_) / 256, blk, 0, stream>>>(cpart, b_qkv, nullptr, qkv, QKV_, 0);
  // flash-decode attention over cached K/V (+ new token in combine)
  attn_partial<<<dim3(B_ * NH_, NCHUNK), blk, 0, stream>>>(qkv, Kc, Vc, part);
  attn_combine<<<B_ * NH_, 128, 0, stream>>>(part, qkv, attn);
  // h = attn @ W_proj + b_proj + hidden
  gemm16_partial<<<dim3(H_ / 128, KSPLIT), blk, 0, stream>>>(attn, W_proj, cpart, H_, H_);
  gemm16_finish<<<(B_ * H_) / 256, blk, 0, stream>>>(cpart, b_proj, hidden, h, H_, 1);
  // y = LN2(h)
  ln_kernel<<<B_, blk, 0, stream>>>(h, ln2_g, ln2_b, y_ln);
  // ff = gelu_new(y @ W_fc + b_fc)
  gemm16_partial<<<dim3(FF_ / 128, KSPLIT), blk, 0, stream>>>(y_ln, W_fc, cpart, H_, FF_);
  gemm16_finish<<<(B_ * FF_) / 256, blk, 0, stream>>>(cpart, b_fc, nullptr, ff, FF_, 2);
  // out = ff @ W_out + b_out + h
  gemm16_partial<<<dim3(H_ / 128, KSPLIT), blk, 0, stream>>>(ff, W_out, cpart, FF_, H_);
  gemm16_finish<<<(B_ * H_) / 256, blk, 0, stream>>>(cpart, b_out, h, out, H_, 1);
}